// CattedLayerLP_45500883534614
// MI455X (gfx1250) — compile-verified
//
#include <hip/hip_runtime.h>
#include <math.h>

typedef float v2f __attribute__((ext_vector_type(2)));
typedef float v8f __attribute__((ext_vector_type(8)));

#define HDIM 64
#define NLAYERS 4

// ---------------- degree / misc ----------------

__global__ void init_kernel(float* __restrict__ deg, float* __restrict__ accum, int N) {
  int i = blockIdx.x * blockDim.x + threadIdx.x;
  if (i < N) deg[i] = 1.0f;            // deg starts at 1 (self term)
  if (i == 0) accum[0] = 0.0f;
}

__global__ void deg_count_kernel(const int* __restrict__ dst, float* __restrict__ deg, int E) {
  int e = blockIdx.x * blockDim.x + threadIdx.x;
  if (e < E)
    __hip_atomic_fetch_add(&deg[dst[e]], 1.0f, __ATOMIC_RELAXED, __HIP_MEMORY_SCOPE_AGENT);
}

__global__ void rsqrt_kernel(float* __restrict__ d, int N) {
  int i = blockIdx.x * blockDim.x + threadIdx.x;
  if (i < N) d[i] = rsqrtf(d[i]);      // dis = 1/sqrt(deg), in place
}

__global__ void zero_kernel(float* __restrict__ p, int n) {
  int i = blockIdx.x * blockDim.x + threadIdx.x;
  if (i < n) p[i] = 0.0f;
}

// ---------------- fp32 WMMA GEMM:  H[N x 64] = X[N x K] @ W[K x 64] ----------------
// One wave per 16x16 output tile. v_wmma_f32_16x16x4_f32, K stepped by 4.
// A (16x4): lane l<16 row M=l holds K=k..k+1 in (a.x,a.y); lanes 16-31 hold K=k+2..k+3.
// B (4x16): symmetric, N = lane&15 striped across lanes.
// C/D: VGPR v -> M = v + 8*(lane>=16), N = lane&15.
__global__ void gemm_wmma_kernel(const float* __restrict__ X, const float* __restrict__ W,
                                 float* __restrict__ Hout, int N, int K) {
  int wave = (blockIdx.x * blockDim.x + threadIdx.x) >> 5;
  int lane = threadIdx.x & 31;
  int row0 = (wave >> 2) * 16;          // 64/16 = 4 column tiles per row band
  int col0 = (wave & 3) * 16;
  if (row0 >= N) return;                // wave-uniform exit: EXEC stays all-1s for WMMA
  int half = lane >> 4;
  int lr   = lane & 15;

  const float* xrow = X + (size_t)(row0 + lr) * K;
  v8f acc = {};
  for (int k = 0; k < K; k += 4) {
    int ka = k + 2 * half;
    v2f a, b;
    a.x = xrow[ka];
    a.y = xrow[ka + 1];
    b.x = W[(size_t)ka * HDIM + col0 + lr];
    b.y = W[(size_t)(ka + 1) * HDIM + col0 + lr];
    acc = __builtin_amdgcn_wmma_f32_16x16x4_f32(false, a, false, b, (short)0, acc, false, false);
  }
#pragma unroll
  for (int v = 0; v < 8; ++v)
    Hout[(size_t)(row0 + v + 8 * half) * HDIM + col0 + lr] = acc[v];
}

// ---------------- message scatter: agg[dst] += h[src] * dis[src]*dis[dst] ----------------
// One thread per (edge, column): a wave covers 32 consecutive columns of one edge,
// so the h[src] read is one contiguous 128B slice and src/dst loads are wave-uniform.
__global__ void scatter_kernel(const float* __restrict__ h, const int* __restrict__ src,
                               const int* __restrict__ dst, const float* __restrict__ dis,
                               float* __restrict__ agg, int E) {
  int g = blockIdx.x * blockDim.x + threadIdx.x;
  int e = g >> 6;
  int c = g & 63;
  if (e >= E) return;
  int s = src[e], d = dst[e];
  float coef = dis[s] * dis[d];
  float val = h[(size_t)s * HDIM + c] * coef;
  __hip_atomic_fetch_add(&agg[(size_t)d * HDIM + c], val,
                         __ATOMIC_RELAXED, __HIP_MEMORY_SCOPE_AGENT);
}

// ---------------- zout = relu(agg + h * dis^2 + b) ----------------
__global__ void finalize_kernel(const float* __restrict__ agg, const float* __restrict__ h,
                                const float* __restrict__ dis, const float* __restrict__ b,
                                float* __restrict__ zout, int n) {
  int g = blockIdx.x * blockDim.x + threadIdx.x;
  if (g >= n) return;
  int i = g >> 6, c = g & 63;
  float di = dis[i];
  float v = agg[g] + h[g] * di * di + b[c];
  zout[g] = fmaxf(v, 0.0f);
}

// ---------------- link prediction + BCE-with-logits accumulation ----------------
// dot[l][m] = sum_h zs[l, e0, h] * zs[m, e1, h]; logit = dot(flat) . Wp + bp.
// target == 1 only for global logits row EP, i.e. edge 0 of the neg set.
__global__ void linkpred_kernel(const float* __restrict__ zs, const int* __restrict__ e0,
                                const int* __restrict__ e1, const float* __restrict__ Wp,
                                const float* __restrict__ bp, float* __restrict__ accum,
                                int EP, int layerStride, int target_j) {
  int j = blockIdx.x * blockDim.x + threadIdx.x;
  float term = 0.0f;
  if (j < EP) {
    const float* sa = zs + (size_t)e0[j] * HDIM;
    const float* sb = zs + (size_t)e1[j] * HDIM;
    float dot[16];
#pragma unroll
    for (int t = 0; t < 16; ++t) dot[t] = 0.0f;
    for (int hh = 0; hh < HDIM; ++hh) {
      float s[4], d[4];
#pragma unroll
      for (int l = 0; l < 4; ++l) {
        s[l] = sa[(size_t)l * layerStride + hh];
        d[l] = sb[(size_t)l * layerStride + hh];
      }
#pragma unroll
      for (int l = 0; l < 4; ++l)
#pragma unroll
        for (int m = 0; m < 4; ++m)
          dot[l * 4 + m] += s[l] * d[m];
    }
    float logit = bp[0];
#pragma unroll
    for (int t = 0; t < 16; ++t) logit += dot[t] * Wp[t];
    float target = (j == target_j) ? 1.0f : 0.0f;
    term = fmaxf(logit, 0.0f) - logit * target + log1pf(expf(-fabsf(logit)));
  }
  // wave32 reduction, then one atomic per wave
#pragma unroll
  for (int o = 16; o > 0; o >>= 1)
    term += __shfl_down(term, o, 32);
  if ((threadIdx.x & 31) == 0)
    __hip_atomic_fetch_add(accum, term, __ATOMIC_RELAXED, __HIP_MEMORY_SCOPE_AGENT);
}

__global__ void mean_kernel(const float* __restrict__ accum, float* __restrict__ out,
                            float inv_count) {
  out[0] = accum[0] * inv_count;
}

// ---------------- driver ----------------

extern "C" void kernel_launch(void* const* d_in, const int* in_sizes, int n_in,
                              void* d_out, int out_size, void* d_ws, size_t ws_size,
                              hipStream_t stream) {
  const float* x   = (const float*)d_in[0];
  const int*   ei  = (const int*)d_in[1];
  const int*   pos = (const int*)d_in[2];
  const int*   neg = (const int*)d_in[3];
  const float* W0  = (const float*)d_in[4];
  const float* b0  = (const float*)d_in[5];
  const float* Wh  = (const float*)d_in[6];
  const float* bh  = (const float*)d_in[7];
  const float* Wp  = (const float*)d_in[8];
  const float* bp  = (const float*)d_in[9];

  const int IN = 128;
  const int N  = in_sizes[0] / IN;   // 50000
  const int E  = in_sizes[1] / 2;    // 800000
  const int EP = in_sizes[2] / 2;    // 100000
  const int nh = N * HDIM;

  // Workspace layout (floats): dis[N] | h[N*64] | agg[N*64] | zs[4*N*64] | accum[1]
  float* ws    = (float*)d_ws;
  float* dis   = ws;
  float* hbuf  = dis  + N;
  float* agg   = hbuf + (size_t)nh;
  float* zs    = agg  + (size_t)nh;
  float* accum = zs   + (size_t)NLAYERS * nh;

  const int* src  = ei;
  const int* dstv = ei + E;

  // symmetric-normalization coefficients
  init_kernel<<<(N + 255) / 256, 256, 0, stream>>>(dis, accum, N);
  deg_count_kernel<<<(E + 255) / 256, 256, 0, stream>>>(dstv, dis, E);
  rsqrt_kernel<<<(N + 255) / 256, 256, 0, stream>>>(dis, N);

  int tiles_m     = (N + 15) / 16;
  int waves       = tiles_m * 4;              // 4 column tiles of the 64-wide output
  int gemm_blocks = (waves + 3) / 4;          // 4 waves (128 threads) per block

  for (int l = 0; l < NLAYERS; ++l) {
    const float* Xin = (l == 0) ? x  : (zs + (size_t)(l - 1) * nh);
    const int    K   = (l == 0) ? IN : HDIM;
    const float* Wl  = (l == 0) ? W0 : (Wh + (size_t)(l - 1) * HDIM * HDIM);
    const float* bl  = (l == 0) ? b0 : (bh + (size_t)(l - 1) * HDIM);

    gemm_wmma_kernel<<<gemm_blocks, 128, 0, stream>>>(Xin, Wl, hbuf, N, K);
    zero_kernel<<<(nh + 255) / 256, 256, 0, stream>>>(agg, nh);
    int sthreads = E * HDIM;
    scatter_kernel<<<(sthreads + 255) / 256, 256, 0, stream>>>(hbuf, src, dstv, dis, agg, E);
    finalize_kernel<<<(nh + 255) / 256, 256, 0, stream>>>(agg, hbuf, dis, bl,
                                                          zs + (size_t)l * nh, nh);
  }

  linkpred_kernel<<<(EP + 255) / 256, 256, 0, stream>>>(zs, pos, pos + EP, Wp, bp,
                                                        accum, EP, nh, -1);
  linkpred_kernel<<<(EP + 255) / 256, 256, 0, stream>>>(zs, neg, neg + EP, Wp, bp,
                                                        accum, EP, nh, 0);
  mean_kernel<<<1, 1, 0, stream>>>(accum, (float*)d_out, 1.0f / (float)(2 * EP));
}